// HDBLUT_87454124081251
// MI455X (gfx1250) — compile-verified
//
#include <hip/hip_runtime.h>
#include <cstdint>

typedef float v4f __attribute__((ext_vector_type(4)));

#define N   2048
#define NO  4096
#define TPB 1024

__device__ __forceinline__ int iabs(int x) { return x < 0 ? -x : x; }

__global__ __launch_bounds__(TPB) void hdblut_fused_kernel(
    const int* __restrict__ img, const float* __restrict__ wgt,
    float* __restrict__ out)
{
    // Full LUT: 3 tables x 4096 entries x float4 = 196608 B (fits 320KB/WGP LDS)
    __shared__ alignas(16) float lut[3 * 4096 * 4];
    const int tid = threadIdx.x;

    // ---- Stage LUT into LDS with CDNA5 async global->LDS copies ----
    {
        unsigned ldsb = (unsigned)(uintptr_t)&lut[0];   // low 32 bits = LDS offset
        const uint64_t src = (uint64_t)wgt;
        #pragma unroll
        for (int it = 0; it < 12; ++it) {               // 12288 x b128 / 1024 thr
            unsigned off = (unsigned)(tid + it * TPB) * 16u;
            unsigned lad = ldsb + off;
            asm volatile("global_load_async_to_lds_b128 %0, %1, %2"
                         :: "v"(lad), "v"(off), "s"(src) : "memory");
        }
        asm volatile("s_wait_asynccnt 0" ::: "memory");
    }
    __syncthreads();

    const v4f* lut4 = (const v4f*)lut;

    const int tpx = tid & 15;        // 16 strips of 4 pixels = 64 cols
    const int tpy = tid >> 4;        // 64 rows
    const int gj0 = blockIdx.x * 64 + tpx * 4;

    constexpr int BP[3] = {0,1,1}, BQ[3] = {1,1,2};
    constexpr int CP[3] = {0,2,2}, CQ[3] = {2,2,1};
    // w4 component s -> acc sub-index, per rotation
    constexpr int P[4][4] = {{0,1,2,3},{1,3,0,2},{3,2,1,0},{2,0,3,1}};

    for (int half = 0; half < 2; ++half) {
        const int gi = blockIdx.y * 128 + half * 64 + tpy;

        int V[5][8];   // rows gi-2..gi+2, cols gj0-2..gj0+5
        const bool interior = (gi >= 2) & (gi < N - 2) & (gj0 >= 2) & (gj0 <= N - 6);
        if (interior) {
            #pragma unroll
            for (int d = 0; d < 5; ++d) {
                const int2* p = (const int2*)(img + (size_t)(gi - 2 + d) * N + (gj0 - 2));
                int2 x0 = p[0], x1 = p[1], x2 = p[2], x3 = p[3];
                V[d][0] = x0.x; V[d][1] = x0.y; V[d][2] = x1.x; V[d][3] = x1.y;
                V[d][4] = x2.x; V[d][5] = x2.y; V[d][6] = x3.x; V[d][7] = x3.y;
            }
        } else {
            #pragma unroll
            for (int d = 0; d < 5; ++d) {
                int m  = gi - 2 + d;
                int ri = (N - 1) - iabs((N - 1) - iabs(m));   // reflect both ends
                const int* rp = img + (size_t)ri * N;
                #pragma unroll
                for (int e = 0; e < 8; ++e) {
                    int n  = gj0 - 2 + e;
                    int rj = (N - 1) - iabs((N - 1) - iabs(n));
                    V[d][e] = rp[rj];
                }
            }
        }

        float acc[4][4];
        #pragma unroll
        for (int m = 0; m < 4; ++m)
            #pragma unroll
            for (int s = 0; s < 4; ++s) acc[m][s] = 0.f;

        #pragma unroll
        for (int r = 0; r < 4; ++r) {
            #pragma unroll
            for (int k = 0; k < 3; ++k) {
                const int bp = BP[k], bq = BQ[k], cp = CP[k], cq = CQ[k];
                int bdi, bdj, cdi, cdj;   // rotated neighbor deltas (orig coords)
                if      (r == 0) { bdi =  bp; bdj =  bq; cdi =  cp; cdj =  cq; }
                else if (r == 1) { bdi =  bq; bdj = -bp; cdi =  cq; cdj = -cp; }
                else if (r == 2) { bdi = -bp; bdj = -bq; cdi = -cp; cdj = -cq; }
                else             { bdi = -bq; bdj =  bp; cdi = -cq; cdj =  cp; }
                #pragma unroll
                for (int m = 0; m < 4; ++m) {
                    const int a = V[2][2 + m];
                    const int b = V[2 + bdi][2 + m + bdj];
                    const int c = V[2 + cdi][2 + m + cdj];
                    const int idx = (a << 8) | (b << 4) | c;
                    v4f w = lut4[(k << 12) | idx];          // ds_load_b128 gather
                    acc[m][P[r][0]] += w.x;
                    acc[m][P[r][1]] += w.y;
                    acc[m][P[r][2]] += w.z;
                    acc[m][P[r][3]] += w.w;
                }
            }
        }

        const float sc = 1.0f / 3.0f;
        size_t o = (size_t)(2 * gi) * NO + 2 * gj0;
        v4f t0 = {acc[0][0]*sc, acc[0][1]*sc, acc[1][0]*sc, acc[1][1]*sc};
        v4f t1 = {acc[2][0]*sc, acc[2][1]*sc, acc[3][0]*sc, acc[3][1]*sc};
        v4f b0 = {acc[0][2]*sc, acc[0][3]*sc, acc[1][2]*sc, acc[1][3]*sc};
        v4f b1 = {acc[2][2]*sc, acc[2][3]*sc, acc[3][2]*sc, acc[3][3]*sc};
        __builtin_nontemporal_store(t0, (v4f*)(out + o));
        __builtin_nontemporal_store(t1, (v4f*)(out + o + 4));
        __builtin_nontemporal_store(b0, (v4f*)(out + o + NO));
        __builtin_nontemporal_store(b1, (v4f*)(out + o + NO + 4));
    }
}

extern "C" void kernel_launch(void* const* d_in, const int* in_sizes, int n_in,
                              void* d_out, int out_size, void* d_ws, size_t ws_size,
                              hipStream_t stream)
{
    (void)in_sizes; (void)n_in; (void)out_size; (void)d_ws; (void)ws_size;
    const int*   img = (const int*)d_in[0];     // 2048x2048 int32, values 0..15
    const float* w   = (const float*)d_in[1];   // 3 x 4096 x 4 float32
    float*       out = (float*)d_out;           // 4096x4096 float32

    dim3 grid(N / 64, N / 128);                 // 64-col x 128-row tiles
    hdblut_fused_kernel<<<grid, TPB, 0, stream>>>(img, w, out);
}